// TKBCModel_66340064854335
// MI455X (gfx1250) — compile-verified
//
#include <hip/hip_runtime.h>

typedef __attribute__((ext_vector_type(2))) float v2f;
typedef __attribute__((ext_vector_type(8))) float v8f;

#define B_   1000
#define D_   512
#define N_   100000
#define F_   50
#define WN   128     // n-columns per block (slab staged in LDS)
#define SRHS 136     // padded LDS row stride for rhs slab (words): 2*136 % 64 = 16 -> halves hit disjoint banks
#define SQ   516     // padded LDS row stride for q tile (words): 4*row % 64 distinct for 16 rows
#define NBT  63      // ceil(B_/16)

// ---------------------------------------------------------------------------
// CDNA5 async global->LDS copy (GLOBAL_LOAD_ASYNC_TO_LDS_B128, ASYNCcnt).
// vdst VGPR carries the LDS byte address = low 32 bits of the generic pointer
// (ISA 10.2).
// ---------------------------------------------------------------------------
__device__ __forceinline__ void async_copy_b128(const float* gsrc, float* ldst) {
    unsigned lofs = (unsigned)(size_t)ldst;
    asm volatile("global_load_async_to_lds_b128 %0, %1, off"
                 :: "v"(lofs), "v"(gsrc) : "memory");
}

__device__ __forceinline__ void wait_async0() {
#if __has_builtin(__builtin_amdgcn_s_wait_asynccnt)
    __builtin_amdgcn_s_wait_asynccnt(0);
#else
    asm volatile("s_wait_asynccnt 0x0" ::: "memory");
#endif
}

// ---------------------------------------------------------------------------
// Kernel 1: per-row target score t_b and filter correction.
//   rank_b = 1 + |{n : score >= t_b}| - |{distinct filtered n : score >= t_b}|
// Pre-store out[b] = 1 - (distinct filtered hits); GEMM kernel adds the count.
// ---------------------------------------------------------------------------
__global__ __launch_bounds__(64) void rank_init_kernel(
    const float* __restrict__ q, const float* __restrict__ rhs,
    const int* __restrict__ fidx, const int* __restrict__ tgt,
    float* __restrict__ t_arr, float* __restrict__ out)
{
    __shared__ float s[F_ + 1];
    __shared__ int   idx[F_ + 1];
    const int b = blockIdx.x;
    const int t = threadIdx.x;

    if (t < F_)  idx[t]  = fidx[b * F_ + t];
    if (t == F_) idx[F_] = tgt[b];
    __syncthreads();

    if (t <= F_) {
        const int n = idx[t];
        const float* qb = q + (size_t)b * D_;
        float acc = 0.f;
        for (int k = 0; k < D_; ++k)
            acc = fmaf(qb[k], rhs[(size_t)k * N_ + n], acc);
        s[t] = acc;
    }
    __syncthreads();

    if (t == 0) {
        const float tv = s[F_];
        t_arr[b] = tv;
        int c = 0;
        for (int j = 0; j <= F_; ++j) {             // de-dup: distinct positions only
            bool dup = false;
            for (int p = 0; p < j; ++p)
                if (idx[p] == idx[j]) { dup = true; break; }
            if (!dup && s[j] >= tv) ++c;
        }
        out[b] = 1.0f - (float)c;
    }
}

// ---------------------------------------------------------------------------
// Kernel 2: fused GEMM + threshold count.  One block per 128-wide n-slab.
// rhs slab async-staged to LDS once (HBM reads rhs exactly once); q tiles
// (16x512) + the tile's 16 t-values staged per btile; 8 waves, one 16-col
// subtile each; K swept with v_wmma_f32_16x16x4_f32; ballot-based counting.
// ---------------------------------------------------------------------------
extern __shared__ float smem[];

__global__ __launch_bounds__(256) void rank_count_kernel(
    const float* __restrict__ q, const float* __restrict__ rhs,
    const float* __restrict__ t_arr, float* __restrict__ out)
{
    float* lds_rhs = smem;                       // [D_][SRHS]
    float* lds_q   = smem + D_ * SRHS;           // [16][SQ]
    float* lds_t   = lds_q + 16 * SQ;            // [16]

    const int t  = threadIdx.x;
    const int n0 = blockIdx.x * WN;

    // ---- async-stage the rhs slab: D_ rows x WN cols, padded row stride
    {
        const int kr = t >> 5;                   // 8 rows per sweep of 256 threads
        const int nl = (t & 31) * 4;             // 16B per lane, coalesced
        const int gn = n0 + nl;
        for (int r0 = 0; r0 < D_; r0 += 8) {
            const int k = r0 + kr;
            const float* src = rhs + (size_t)k * N_ + gn;
            float* dst = &lds_rhs[k * SRHS + nl];
            if (gn + 3 < N_) {
                async_copy_b128(src, dst);
            } else {                              // only in the last slab
                float4 v;
                v.x = (gn + 0 < N_) ? src[0] : 0.f;
                v.y = (gn + 1 < N_) ? src[1] : 0.f;
                v.z = (gn + 2 < N_) ? src[2] : 0.f;
                v.w = (gn + 3 < N_) ? src[3] : 0.f;
                *reinterpret_cast<float4*>(dst) = v;
            }
        }
        wait_async0();                            // own async writes done pre-barrier
    }

    const int w      = t >> 5;                   // wave id -> n-subtile
    const int lane   = t & 31;
    const int lo     = lane & 15;
    const int hi     = lane >> 4;
    const int nlocal = w * 16 + lo;
    const int gn     = n0 + nlocal;
    const unsigned vmask = (unsigned)__ballot(gn < N_);   // loop-invariant lane mask

    for (int bt = 0; bt < NBT; ++bt) {
        __syncthreads();                         // slab ready / prev q tile consumed

        // ---- stage q tile (16 x D_) + this tile's 16 threshold values
        {
            const int row = t >> 4;
            const int c0  = (t & 15) * 4;
            const int gb  = bt * 16 + row;
#pragma unroll
            for (int j = 0; j < 8; ++j) {
                const int col = c0 + j * 64;
                if (gb < B_) {
                    async_copy_b128(q + (size_t)gb * D_ + col, &lds_q[row * SQ + col]);
                } else {
                    *reinterpret_cast<float4*>(&lds_q[row * SQ + col]) =
                        make_float4(0.f, 0.f, 0.f, 0.f);
                }
            }
            if (t < 16) {
                const int gbt = bt * 16 + t;
                lds_t[t] = t_arr[gbt < B_ ? gbt : B_ - 1];
            }
            wait_async0();
        }
        __syncthreads();

        // ---- K sweep: 128 x v_wmma_f32_16x16x4_f32 accumulation
        v8f c = {};
        const float* qrow = &lds_q[lo * SQ];
#pragma unroll 4
        for (int k0 = 0; k0 < D_; k0 += 4) {
            const int kb = k0 + 2 * hi;          // A 16x4: lanes 16-31 carry K+2..K+3
            v2f a = *reinterpret_cast<const v2f*>(qrow + kb);
            v2f bv;
            bv.x = lds_rhs[kb * SRHS + nlocal];
            bv.y = lds_rhs[(kb + 1) * SRHS + nlocal];
            c = __builtin_amdgcn_wmma_f32_16x16x4_f32(
                    false, a, false, bv, (short)0, c, false, false);
        }

        // ---- branchless count: ballot per C VGPR (bits 0-15 -> row r,
        //      bits 16-31 -> row r+8); column/row validity via uniform masks
        const int bbase = bt * 16 + 8 * hi;      // C layout: VGPR r -> row r (+8 hi half)
        const float4 ta = *reinterpret_cast<const float4*>(&lds_t[8 * hi]);
        const float4 tb = *reinterpret_cast<const float4*>(&lds_t[8 * hi + 4]);
        const float tv[8] = {ta.x, ta.y, ta.z, ta.w, tb.x, tb.y, tb.z, tb.w};
        float cl[8], ch[8];                      // wave-uniform per-row counts
#pragma unroll
        for (int r = 0; r < 8; ++r) {
            const unsigned m = ((unsigned)__ballot(c[r] >= tv[r])) & vmask;
            cl[r] = (float)__popc(m & 0xFFFFu);
            ch[r] = (float)__popc(m >> 16);
        }
        if (lo == 0) {                           // lanes 0 and 16 active
#pragma unroll
            for (int r = 0; r < 8; ++r) {
                const float cnt = hi ? ch[r] : cl[r];
                const int   gb  = bbase + r;
                if (cnt > 0.f && gb < B_)        // padded rows rejected here
                    atomicAdd(&out[gb], cnt);    // exact integer-valued f32
            }
        }
    }
}

// ---------------------------------------------------------------------------
extern "C" void kernel_launch(void* const* d_in, const int* in_sizes, int n_in,
                              void* d_out, int out_size, void* d_ws, size_t ws_size,
                              hipStream_t stream) {
    (void)in_sizes; (void)n_in; (void)out_size; (void)ws_size;
    const float* q    = (const float*)d_in[0];
    const float* rhs  = (const float*)d_in[1];
    const int*   fidx = (const int*)d_in[2];
    const int*   tgt  = (const int*)d_in[3];
    float* out   = (float*)d_out;
    float* t_arr = (float*)d_ws;                 // B_ floats of scratch

    rank_init_kernel<<<B_, 64, 0, stream>>>(q, rhs, fidx, tgt, t_arr, out);

    const size_t smem_bytes =
        (size_t)(D_ * SRHS + 16 * SQ + 16) * sizeof(float);  // ~304 KB < 320 KB/WGP
    (void)hipFuncSetAttribute((const void*)rank_count_kernel,
                              hipFuncAttributeMaxDynamicSharedMemorySize,
                              (int)smem_bytes);
    const int nblocks = (N_ + WN - 1) / WN;      // 782 slabs
    rank_count_kernel<<<nblocks, 256, smem_bytes, stream>>>(q, rhs, t_arr, out);
}